// InformedRayStepSampler_43516608643396
// MI455X (gfx1250) — compile-verified
//
#include <hip/hip_runtime.h>
#include <hip/hip_bf16.h>
#include <math.h>

// Informed ray-step sampler (NeRF-style inverse-CDF resampling).
// N_RAYS rays; 64 coarse samples -> 65-entry CDF; 256 fine samples, sorted,
// emitted transposed (256, N). Memory-bound: ~580 MB @ 23.3 TB/s ~= 25 us.
//
// CDNA5 paths used:
//  - wave32 __shfl scans for the 64-long cumsum (2 elems/lane)
//  - GLOBAL_LOAD_ASYNC_TO_LDS_B32 + s_wait_asynccnt for density staging
//  - global_prefetch_b8 on the streamed u rows
//  - B128 vectorized u loads and transposed output stores
//  - padded LDS (stride 33 / 66 / 257) for conflict-free tile reads,
//    bitonic sort, and the transposed coalesced output.

#define N_COARSE 64
#define N_SAMP   256
#define EPSV     1e-5f
#define TFIN     1e10f

#if __has_builtin(__builtin_amdgcn_global_load_async_to_lds_b32) && \
    __has_builtin(__builtin_amdgcn_s_wait_asynccnt)
#define USE_ASYNC_LDS 1
typedef __attribute__((address_space(1))) int async_gint_t;  // global src
typedef __attribute__((address_space(3))) int async_lint_t;  // LDS dst
#else
#define USE_ASYNC_LDS 0
#endif

__device__ __forceinline__ float edge_at(int j, float tn, float tf) {
  // edges = [tnear, midpoints(ts), tfar], ts[j] = tn + (tf-tn)*(j/63)
  if (j <= 0) return tn;
  if (j >= N_COARSE) return tf;
  float dr = tf - tn;
  float a = tn + dr * ((float)(j - 1) / 63.0f);
  float b = tn + dr * ((float)j / 63.0f);
  return 0.5f * (a + b);
}

__device__ __forceinline__ float wave_incl_scan(float x, int ln) {
#pragma unroll
  for (int d = 1; d < 32; d <<= 1) {
    float y = __shfl_up(x, d, 32);
    if (ln >= d) x += y;
  }
  return x;
}

__global__ __launch_bounds__(1024) void informed_sampler_kernel(
    const float* __restrict__ tnear, const float* __restrict__ tfar,
    const float* __restrict__ dnorm, const float* __restrict__ density,
    const float* __restrict__ u, float* __restrict__ out, int N) {
  __shared__ float C[32 * 66];   // per-ray cdf[0..64]   (stride 66, padded)
  __shared__ float E[32 * 66];   // per-ray edges[0..64] (stride 66, padded)
  __shared__ float S[32 * 257];  // sort buffer (stride 257); phase A: 64x32
                                 // density tile at stride 33 (2112 floats)

  const int tid = threadIdx.x;
  const int wv  = tid >> 5;   // wave index = ray-in-block
  const int ln  = tid & 31;   // lane (wave32)
  const int r0  = blockIdx.x * 32;
  const int r   = min(r0 + wv, N - 1);

  if (tid < 32)
    __builtin_prefetch(u + (size_t)min(r0 + tid, N - 1) * N_SAMP, 0, 1);

  // ---- Phase A0: stage density tile (t=0..63) x (32 rays), coalesced ----
#pragma unroll
  for (int c = 0; c < 2; ++c) {
    int e = tid + 1024 * c;          // 0..2047
    int t = e >> 5, rib = e & 31;
    int rr = min(r0 + rib, N - 1);
    const float* gp = density + (size_t)t * N + rr;
#if USE_ASYNC_LDS
    __builtin_amdgcn_global_load_async_to_lds_b32(
        (async_gint_t*)gp, (async_lint_t*)&S[t * 33 + rib], 0, 0);
#else
    S[t * 33 + rib] = *gp;
#endif
  }
#if USE_ASYNC_LDS
  __builtin_amdgcn_s_wait_asynccnt(0);
#endif
  __syncthreads();

  const float tn = tnear[r];
  const float tf = tfar[r];
  const float dn = dnorm[r];
  const float dr = tf - tn;

  // ---- Phase A1: sd = density * delta * dnorm for t = ln, ln+32 ----
  float sdv[2];
#pragma unroll
  for (int c = 0; c < 2; ++c) {
    int t = ln + 32 * c;
    float ts0 = tn + dr * ((float)t / 63.0f);
    float del;
    if (t < 63) {
      float ts1 = tn + dr * ((float)(t + 1) / 63.0f);
      del = ts1 - ts0;
    } else {
      del = TFIN - ts0;               // sentinel far-plane delta
    }
    float d = S[t * 33 + wv];         // conflict-free (pad 33)
    sdv[c] = d * del * dn;
  }

  // exclusive cumsum of sd over 64 (two 32-chunks, wave scan)
  float i0   = wave_incl_scan(sdv[0], ln);
  float tot0 = __shfl(i0, 31, 32);
  float i1   = wave_incl_scan(sdv[1], ln) + tot0;
  float ex0 = i0 - sdv[0];
  float ex1 = i1 - sdv[1];

  // weights = T * (1 - exp(-sd)); w = weights + EPS
  float w0 = __expf(-ex0) * (1.0f - __expf(-sdv[0])) + EPSV;
  float w1 = __expf(-ex1) * (1.0f - __expf(-sdv[1])) + EPSV;

  // inclusive prefix of w -> normalized CDF
  float W0  = wave_incl_scan(w0, ln);
  float tW0 = __shfl(W0, 31, 32);
  float W1  = wave_incl_scan(w1, ln) + tW0;
  float total = __shfl(W1, 31, 32);
  float inv = 1.0f / total;

  float* Cw = &C[wv * 66];
  float* Ew = &E[wv * 66];
  if (ln == 0) Cw[0] = 0.0f;
  Cw[ln + 1]  = W0 * inv;
  Cw[ln + 33] = W1 * inv;
#pragma unroll
  for (int j = ln; j < 65; j += 32) Ew[j] = edge_at(j, tn, tf);

  __syncthreads();  // density tile fully consumed; S becomes sort buffer

  // ---- Phase B: inverse-CDF sample 256 u's (2x float4 per lane) ----
  float* Sw = &S[wv * 257];
  const float4* u4 = reinterpret_cast<const float4*>(u + (size_t)r * N_SAMP);
#pragma unroll
  for (int half = 0; half < 2; ++half) {
    float4 uu4 = u4[ln + 32 * half];  // samples s = 128*half + 4*ln .. +3
#pragma unroll
    for (int q = 0; q < 4; ++q) {
      int s = 128 * half + 4 * ln + q;
      float uu = (q == 0) ? uu4.x : (q == 1) ? uu4.y : (q == 2) ? uu4.z : uu4.w;
      // searchsorted(cdf, u, side='right') over Cw[0..64] -> idx in [1,64]
      int lo = 0, hi = 65;
#pragma unroll
      for (int it = 0; it < 7; ++it) {
        if (hi - lo > 1) {
          int mid = (lo + hi) >> 1;
          if (Cw[mid] > uu) hi = mid; else lo = mid;
        }
      }
      int idx = hi;
      if (idx > N_COARSE) idx = N_COARSE;
      if (idx < 1) idx = 1;
      int below = idx - 1;
      float cb = Cw[below], ca = Cw[idx];
      float eb = Ew[below], ea = Ew[idx];
      float denom = ca - cb;
      denom = (denom < EPSV) ? 1.0f : denom;
      float t = (uu - cb) / denom;
      Sw[s] = eb + t * (ea - eb);
    }
  }

  // ---- Phase C: block-cooperative bitonic sort of 32 segments of 256 ----
  for (int k = 2; k <= 256; k <<= 1) {
    for (int jj = k >> 1; jj > 0; jj >>= 1) {
      __syncthreads();
#pragma unroll
      for (int pp = 0; pp < 4; ++pp) {
        int p = tid + 1024 * pp;                 // 0..4095 pairs
        int seg = p >> 7;
        int q = p & 127;
        int i = ((q & ~(jj - 1)) << 1) | (q & (jj - 1));
        int base = seg * 257;
        int ia = base + i, ib = base + (i | jj);
        float a = S[ia], b = S[ib];
        bool asc = ((i & k) == 0);
        bool sw = asc ? (a > b) : (a < b);
        if (sw) { S[ia] = b; S[ib] = a; }
      }
    }
  }
  __syncthreads();

  // ---- Phase D: transposed write-out out[s*N + ray], B128 per thread ----
  if ((N & 3) == 0) {
#pragma unroll
    for (int p = 0; p < 2; ++p) {
      int q = tid + 1024 * p;     // quad index 0..2047
      int s = q >> 3;
      int rib = (q & 7) * 4;
      if (r0 + rib + 3 < N) {
        float4 v;                  // pad-257 gather: bank-conflict-free
        v.x = S[(rib + 0) * 257 + s];
        v.y = S[(rib + 1) * 257 + s];
        v.z = S[(rib + 2) * 257 + s];
        v.w = S[(rib + 3) * 257 + s];
        *reinterpret_cast<float4*>(out + (size_t)s * N + (r0 + rib)) = v;
      }
    }
  } else {
#pragma unroll
    for (int jo = 0; jo < 8; ++jo) {
      int e = tid + 1024 * jo;  // 0..8191
      int s = e >> 5, rib = e & 31;
      if (r0 + rib < N)
        out[(size_t)s * N + (r0 + rib)] = S[rib * 257 + s];
    }
  }
}

extern "C" void kernel_launch(void* const* d_in, const int* in_sizes, int n_in,
                              void* d_out, int out_size, void* d_ws, size_t ws_size,
                              hipStream_t stream) {
  (void)n_in; (void)out_size; (void)d_ws; (void)ws_size;
  const float* tnear   = (const float*)d_in[0];
  const float* tfar    = (const float*)d_in[1];
  const float* dnorm   = (const float*)d_in[2];
  const float* density = (const float*)d_in[3];
  const float* u       = (const float*)d_in[4];
  float* out = (float*)d_out;
  int N = in_sizes[0];                 // 262144
  int blocks = (N + 31) / 32;          // 32 rays per 1024-thread block
  hipLaunchKernelGGL(informed_sampler_kernel, dim3(blocks), dim3(1024), 0,
                     stream, tnear, tfar, dnorm, density, u, out, N);
}